// SetTokenTeacher_30485677867483
// MI455X (gfx1250) — compile-verified
//
#include <hip/hip_runtime.h>

// MI455X / gfx1250, wave32. Fused MLP + dual cross-attention, one WG per (b,t).
// All matmuls on V_WMMA_F32_16X16X4_F32 (fp32-exact vs reference).
// MLP GEMMs register-blocked 4(M) x 2(N): 8 WMMAs per 4 LDS + 4 global loads.

typedef __attribute__((ext_vector_type(2))) float v2f;
typedef __attribute__((ext_vector_type(8))) float v8f;

#define Bdim 16
#define Tdim 64
#define Ndim 64
#define OBSdim 128
#define H1dim 512
#define H2dim 512
#define Ddim 256
#define Kdim 16

// LDS strides padded by 4 floats (64 banks, conflict-free column walks)
#define SX_STRIDE 132   // obs tile 64x132
#define SH_STRIDE 516   // h1/h2 64x516
#define SE_STRIDE 260   // emb 64x260, tokens 16x260
#define SC_STRIDE 68    // scores 16x68
#define SC2_STRIDE 20   // scores2 64x20

// float offsets into dynamic LDS
#define OFF_X 0
#define OFF_TOK 0                                   // overlays dead obs region
#define OFF_SC (OFF_TOK + Kdim * SE_STRIDE)         // 4160
#define OFF_SC2 (OFF_SC + Kdim * SC_STRIDE)         // 5248 (+1280 = 6528 <= 8448)
#define REGION0 (Ndim * SX_STRIDE)                  // 8448
#define OFF_H1 REGION0
#define OFF_EMB REGION0                             // overlays dead h1
#define OFF_H2 (REGION0 + Ndim * SH_STRIDE)         // 41472
#define LDS_FLOATS (OFF_H2 + Ndim * SH_STRIDE)      // 74496 floats = 291 KB

__device__ __forceinline__ v8f wmma_f32(v2f a, v2f b, v8f c) {
  return __builtin_amdgcn_wmma_f32_16x16x4_f32(false, a, false, b, (short)0, c,
                                               false, false);
}

// C[64, NTILES*16] = act(A_lds[64, KD] @ W_glb[KD, WLD] + bias), W row-major.
// Register blocking: 4 M-tiles x 2 N-tiles per accumulator group; B fragments
// reused across the 4 row tiles, A fragments across the 2 column tiles.
template <int NTILES, int KD, int WLD, bool RELU>
__device__ __forceinline__ void mlp_gemm(const float* __restrict__ lin, int istride,
                                         const float* __restrict__ W,
                                         const float* __restrict__ bias,
                                         float* __restrict__ lout, int ostride,
                                         int lane, int wave) {
  const int half = lane >> 4, lcol = lane & 15;
  constexpr int PAIRS = NTILES / 16;  // column-tile pairs per wave
#pragma unroll
  for (int p = 0; p < PAIRS; ++p) {
    const int n0a = (wave + 16 * p) * 16;  // cols: wave+16p and wave+16p+8 tiles
    const int n0b = n0a + 128;
    v8f acc[4][2] = {};
    const float* a0 = lin + lcol * istride + 2 * half;   // A: M=lcol(+16*mi), K=2*half+v
    const float* wb = W + (size_t)(2 * half) * WLD + lcol; // B: K=2*half+v, N=n0+lcol
#pragma unroll 2
    for (int k0 = 0; k0 < KD; k0 += 4) {
      v2f bA, bB;
      bA.x = wb[(size_t)k0 * WLD + n0a];       bA.y = wb[(size_t)(k0 + 1) * WLD + n0a];
      bB.x = wb[(size_t)k0 * WLD + n0b];       bB.y = wb[(size_t)(k0 + 1) * WLD + n0b];
      v2f a[4];
#pragma unroll
      for (int mi = 0; mi < 4; ++mi) {
        const float* ap = a0 + (mi * 16) * istride + k0;
        a[mi].x = ap[0]; a[mi].y = ap[1];
      }
#pragma unroll
      for (int mi = 0; mi < 4; ++mi) {
        acc[mi][0] = wmma_f32(a[mi], bA, acc[mi][0]);
        acc[mi][1] = wmma_f32(a[mi], bB, acc[mi][1]);
      }
    }
    const float bvA = bias[n0a + lcol];
    const float bvB = bias[n0b + lcol];
#pragma unroll
    for (int mi = 0; mi < 4; ++mi) {
#pragma unroll
      for (int r = 0; r < 8; ++r) {           // D: M = 16*mi + r + 8*half
        const int row = mi * 16 + r + 8 * half;
        float va = acc[mi][0][r] + bvA;
        float vb = acc[mi][1][r] + bvB;
        if (RELU) { va = fmaxf(va, 0.0f); vb = fmaxf(vb, 0.0f); }
        lout[row * ostride + n0a + lcol] = va;
        lout[row * ostride + n0b + lcol] = vb;
      }
    }
  }
}

__global__ __launch_bounds__(256) void fused_token_teacher(
    const float* __restrict__ obs, const int* __restrict__ am,
    const float* __restrict__ W1, const float* __restrict__ b1,
    const float* __restrict__ W2, const float* __restrict__ b2,
    const float* __restrict__ W3, const float* __restrict__ b3,
    const float* __restrict__ tq, float* __restrict__ out_tokens,
    float* __restrict__ out_ctx) {
  extern __shared__ float lds[];
  const int bt = blockIdx.x;                 // 0..1023 = b*T + t
  const int tid = threadIdx.x;
  const int lane = tid & 31, wave = tid >> 5;
  const int half = lane >> 4, lcol = lane & 15;
  const int* amrow = am + bt * Ndim;

  // ---- Stage obs tile [64,128] into LDS (coalesced) ----
  const float* obs_bt = obs + (size_t)bt * Ndim * OBSdim;
  for (int i = tid; i < Ndim * OBSdim; i += 256) {
    const int m = i >> 7, k = i & 127;
    lds[OFF_X + m * SX_STRIDE + k] = obs_bt[i];
  }
  __syncthreads();

  // ---- MLP: h1 = relu(X W1 + b1); h2 = relu(h1 W2 + b2); emb = h2 W3 + b3 ----
  mlp_gemm<32, OBSdim, H1dim, true>(lds + OFF_X, SX_STRIDE, W1, b1,
                                    lds + OFF_H1, SH_STRIDE, lane, wave);
  __syncthreads();
  mlp_gemm<32, H1dim, H2dim, true>(lds + OFF_H1, SH_STRIDE, W2, b2,
                                   lds + OFF_H2, SH_STRIDE, lane, wave);
  __syncthreads();
  mlp_gemm<16, H2dim, Ddim, false>(lds + OFF_H2, SH_STRIDE, W3, b3,
                                   lds + OFF_EMB, SE_STRIDE, lane, wave);
  __syncthreads();

  // ---- Attention 1: scores[16,64] = tq @ emb^T / 16 (WMMA, waves 0..3) ----
  if (wave < 4) {
    const int n0 = wave * 16;
    v8f acc = {};
    const float* arow = tq + lcol * Ddim + 2 * half;
    const float* brow = lds + OFF_EMB + (n0 + lcol) * SE_STRIDE + 2 * half; // emb^T
#pragma unroll 4
    for (int k0 = 0; k0 < Ddim; k0 += 4) {
      v2f a; a.x = arow[k0]; a.y = arow[k0 + 1];
      v2f b; b.x = brow[k0]; b.y = brow[k0 + 1];
      acc = wmma_f32(a, b, acc);
    }
#pragma unroll
    for (int r = 0; r < 8; ++r)
      lds[OFF_SC + (r + 8 * half) * SC_STRIDE + n0 + lcol] = acc[r] * 0.0625f;
  }
  __syncthreads();

  // masked softmax over M=64 (matches jnp.where(-1e9)/clip(1e-8) semantics)
  if (tid < Kdim) {
    float* srow = lds + OFF_SC + tid * SC_STRIDE;
    float mx = -1e30f;
    for (int m = 0; m < Ndim; ++m)
      if (amrow[m]) mx = fmaxf(mx, srow[m]);
    float sum = 0.0f;
    for (int m = 0; m < Ndim; ++m) {
      const float w = amrow[m] ? expf(srow[m] - mx) : 0.0f;
      srow[m] = w; sum += w;
    }
    const float inv = 1.0f / fmaxf(sum, 1e-8f);
    for (int m = 0; m < Ndim; ++m) srow[m] *= inv;
  }
  __syncthreads();

  // ---- tokens[16,256] = weights[16,64] @ emb[64,256] ----
  for (int ti = wave; ti < 16; ti += 8) {
    const int n0 = ti * 16;
    v8f acc = {};
    const float* arow = lds + OFF_SC + lcol * SC_STRIDE + 2 * half;
    const float* bcol = lds + OFF_EMB + (2 * half) * SE_STRIDE + n0 + lcol;
#pragma unroll
    for (int k0 = 0; k0 < Ndim; k0 += 4) {
      v2f a; a.x = arow[k0]; a.y = arow[k0 + 1];
      v2f b; b.x = bcol[k0 * SE_STRIDE]; b.y = bcol[(k0 + 1) * SE_STRIDE];
      acc = wmma_f32(a, b, acc);
    }
#pragma unroll
    for (int r = 0; r < 8; ++r) {
      const int m = r + 8 * half;
      const float v = acc[r];
      lds[OFF_TOK + m * SE_STRIDE + n0 + lcol] = v;
      out_tokens[((size_t)bt * Kdim + m) * Ddim + n0 + lcol] = v;
    }
  }
  __syncthreads();

  // ---- Attention 2: scores2[64,16] = emb @ tokens^T / 16 (waves 0..3) ----
  if (wave < 4) {
    const int m0 = wave * 16;
    v8f acc = {};
    const float* arow = lds + OFF_EMB + (m0 + lcol) * SE_STRIDE + 2 * half;
    const float* brow = lds + OFF_TOK + lcol * SE_STRIDE + 2 * half; // tokens^T
#pragma unroll 4
    for (int k0 = 0; k0 < Ddim; k0 += 4) {
      v2f a; a.x = arow[k0]; a.y = arow[k0 + 1];
      v2f b; b.x = brow[k0]; b.y = brow[k0 + 1];
      acc = wmma_f32(a, b, acc);
    }
#pragma unroll
    for (int r = 0; r < 8; ++r)
      lds[OFF_SC2 + (m0 + r + 8 * half) * SC2_STRIDE + lcol] = acc[r] * 0.0625f;
  }
  __syncthreads();

  // plain softmax over 16 (no mask), one row per thread
  if (tid < Ndim) {
    float* srow = lds + OFF_SC2 + tid * SC2_STRIDE;
    float mx = srow[0];
    for (int j = 1; j < Kdim; ++j) mx = fmaxf(mx, srow[j]);
    float sum = 0.0f;
    for (int j = 0; j < Kdim; ++j) { const float w = expf(srow[j] - mx); srow[j] = w; sum += w; }
    const float inv = 1.0f / sum;
    for (int j = 0; j < Kdim; ++j) srow[j] *= inv;
  }
  __syncthreads();

  // ---- ctx[64,256] = weights2[64,16] @ tokens[16,256], masked rows ----
  for (int ti = wave; ti < 64; ti += 8) {
    const int m0 = (ti >> 4) * 16;
    const int n0 = (ti & 15) * 16;
    v8f acc = {};
    const float* arow = lds + OFF_SC2 + (m0 + lcol) * SC2_STRIDE + 2 * half;
    const float* bcol = lds + OFF_TOK + (2 * half) * SE_STRIDE + n0 + lcol;
#pragma unroll
    for (int k0 = 0; k0 < Kdim; k0 += 4) {
      v2f a; a.x = arow[k0]; a.y = arow[k0 + 1];
      v2f b; b.x = bcol[k0 * SE_STRIDE]; b.y = bcol[(k0 + 1) * SE_STRIDE];
      acc = wmma_f32(a, b, acc);
    }
#pragma unroll
    for (int r = 0; r < 8; ++r) {
      const int m = m0 + r + 8 * half;
      const float maskv = amrow[m] ? 1.0f : 0.0f;
      out_ctx[((size_t)bt * Ndim + m) * Ddim + n0 + lcol] = acc[r] * maskv;
    }
  }
}

extern "C" void kernel_launch(void* const* d_in, const int* in_sizes, int n_in,
                              void* d_out, int out_size, void* d_ws, size_t ws_size,
                              hipStream_t stream) {
  (void)in_sizes; (void)n_in; (void)d_ws; (void)ws_size; (void)out_size;
  const float* obs = (const float*)d_in[0];
  const int*   am  = (const int*)d_in[1];
  const float* W1  = (const float*)d_in[2];
  const float* b1  = (const float*)d_in[3];
  const float* W2  = (const float*)d_in[4];
  const float* b2  = (const float*)d_in[5];
  const float* W3  = (const float*)d_in[6];
  const float* b3  = (const float*)d_in[7];
  const float* tq  = (const float*)d_in[8];
  float* out_tokens = (float*)d_out;
  float* out_ctx    = out_tokens + (size_t)Bdim * Tdim * Kdim * Ddim;

  const size_t lds_bytes = (size_t)LDS_FLOATS * sizeof(float);
  // allow >default dynamic LDS (WGP supports 320KB); harmless if already allowed
  (void)hipFuncSetAttribute((const void*)fused_token_teacher,
                            hipFuncAttributeMaxDynamicSharedMemorySize,
                            (int)lds_bytes);

  fused_token_teacher<<<Bdim * Tdim, 256, lds_bytes, stream>>>(
      obs, am, W1, b1, W2, b2, W3, b3, tq, out_tokens, out_ctx);
}